// VSSBlock_88570815578777
// MI455X (gfx1250) — compile-verified
//
#include <hip/hip_runtime.h>
#include <math.h>

#define D_MODEL 384
#define D_STATE 16
#define D_INNER 768
#define DT_RANK 24
#define BATCH   4
#define SEQ_L   4096
#define HWDIM   64
#define M_TOK   (BATCH * SEQ_L)      /* 16384 tokens */
#define N_UZ    (2 * D_INNER)        /* 1536 */
#define XDBL_LD 64                   /* 56 padded to 64 */

typedef __attribute__((ext_vector_type(2))) float v2f;
typedef __attribute__((ext_vector_type(8))) float v8f;

__device__ __forceinline__ float silu_f(float x)     { return x / (1.0f + __expf(-x)); }
__device__ __forceinline__ float softplus_f(float x) { return (x > 20.0f) ? x : log1pf(__expf(x)); }

// ---------------------------------------------------------------------------
// fp32 WMMA register-blocked strip: 2 M-tiles x NT N-tiles per wave.
//   acc[i][j] += A[i*16 : i*16+16, 0:K] * B[0:K, j*16 : j*16+16]
// A frag (16x4 f32): lane l -> M = l%16 ; VGPR v -> K = k0 + v + 2*(l/16)
// B frag (4x16 f32): lane l -> N = l%16 ; VGPR v -> K = k0 + v + 2*(l/16)
// GUARD: B columns >= nvalid contribute 0 (N padding for x_proj's N=56).
// lda/ka are even -> v2f (b64) loads for A are 8-byte aligned.
// ---------------------------------------------------------------------------
template <int NT, bool GUARD>
__device__ __forceinline__ void wmma_block(const float* __restrict__ A, int lda,
                                           const float* __restrict__ B, int ldb,
                                           int K, int nvalid, v8f (&acc)[2][NT]) {
  const int lane = threadIdx.x & 31;
  const int half = lane >> 4;   // 0: lanes 0-15, 1: lanes 16-31
  const int mr   = lane & 15;   // row of A / col of B within a tile
  for (int k = 0; k < K; k += 4) {
    const int ka = k + (half << 1);
    v2f a[2];
    a[0] = *(const v2f*)(A + mr * lda + ka);
    a[1] = *(const v2f*)(A + (16 + mr) * lda + ka);
    v2f b[NT];
#pragma unroll
    for (int j = 0; j < NT; ++j) {
      const int col = j * 16 + mr;
      const bool ok = (!GUARD) || (col < nvalid);
      b[j].x = ok ? B[ka * ldb + col] : 0.0f;
      b[j].y = ok ? B[(ka + 1) * ldb + col] : 0.0f;
    }
#pragma unroll
    for (int i = 0; i < 2; ++i)
#pragma unroll
      for (int j = 0; j < NT; ++j)
        acc[i][j] = __builtin_amdgcn_wmma_f32_16x16x4_f32(false, a[i], false, b[j],
                                                          (short)0, acc[i][j], false, false);
  }
}

// C/D 16x16 f32 layout: VGPR r -> row r + 8*(lane/16), col lane%16
template <int NT>
__device__ __forceinline__ void store_block(float* __restrict__ C, int ldc, v8f (&acc)[2][NT]) {
  const int lane = threadIdx.x & 31;
  const int half = lane >> 4;
  const int mr   = lane & 15;
#pragma unroll
  for (int i = 0; i < 2; ++i)
#pragma unroll
    for (int j = 0; j < NT; ++j)
#pragma unroll
      for (int r = 0; r < 8; ++r)
        C[(i * 16 + r + 8 * half) * ldc + j * 16 + mr] = acc[i][j][r];
}

// ---------------- Kernel 1: u_z = x @ in_proj_w  [16384,384]x[384,1536] ----
__global__ __launch_bounds__(256) void k_in_proj(const float* __restrict__ x,
                                                 const float* __restrict__ w,
                                                 float* __restrict__ uz) {
  const int stripsN = N_UZ / 64;                        // 24
  const int job = blockIdx.x * 8 + (threadIdx.x >> 5);
  const int sm = job / stripsN, sn = job % stripsN;
  v8f acc[2][4] = {};
  wmma_block<4, false>(x + sm * 32 * D_MODEL, D_MODEL, w + sn * 64, N_UZ, D_MODEL, 64, acc);
  store_block<4>(uz + sm * 32 * N_UZ + sn * 64, N_UZ, acc);
}

// ---------------- Kernel 2: depthwise 3x3 conv + SiLU ----------------------
__global__ __launch_bounds__(256) void k_conv(const float* __restrict__ uz,
                                              const float* __restrict__ cw,
                                              const float* __restrict__ cb,
                                              float* __restrict__ u) {
  const int idx = blockIdx.x * 256 + threadIdx.x;       // (b,l,d) flat
  const int d = idx % D_INNER;
  const int l = (idx / D_INNER) % SEQ_L;
  const int b = idx / (D_INNER * SEQ_L);
  const int h = l >> 6, w = l & 63;
  float acc = cb[d];
#pragma unroll
  for (int kh = 0; kh < 3; ++kh) {
    const int hh = h + kh - 1;
    if ((unsigned)hh < HWDIM) {
#pragma unroll
      for (int kw = 0; kw < 3; ++kw) {
        const int ww = w + kw - 1;
        if ((unsigned)ww < HWDIM)
          acc += uz[(b * SEQ_L + (hh << 6) + ww) * N_UZ + d] * cw[d * 9 + kh * 3 + kw];
      }
    }
  }
  u[idx] = silu_f(acc);
}

// ---------------- Kernel 3: x_dbl = u @ x_proj_w  (N=56 -> pad 64) ---------
__global__ __launch_bounds__(256) void k_xproj(const float* __restrict__ u,
                                               const float* __restrict__ w,
                                               float* __restrict__ xdbl) {
  const int sm = blockIdx.x * 8 + (threadIdx.x >> 5);   // 512 M-strips, 1 N-strip
  v8f acc[2][4] = {};
  wmma_block<4, true>(u + sm * 32 * D_INNER, D_INNER, w, DT_RANK + 2 * D_STATE,
                      D_INNER, DT_RANK + 2 * D_STATE, acc);
  store_block<4>(xdbl + sm * 32 * XDBL_LD, XDBL_LD, acc);
}

// ---------------- Kernel 4: delta = softplus(x_dbl[:, :24] @ dt_w + dt_b) --
__global__ __launch_bounds__(256) void k_dt(const float* __restrict__ xdbl,
                                            const float* __restrict__ w,
                                            const float* __restrict__ bias,
                                            float* __restrict__ delta) {
  const int stripsN = D_INNER / 64;                     // 12
  const int job = blockIdx.x * 8 + (threadIdx.x >> 5);
  const int sm = job / stripsN, sn = job % stripsN;
  v8f acc[2][4] = {};
  wmma_block<4, false>(xdbl + sm * 32 * XDBL_LD, XDBL_LD, w + sn * 64, D_INNER, DT_RANK, 64, acc);
  const int lane = threadIdx.x & 31;
  const int half = lane >> 4, mr = lane & 15;
#pragma unroll
  for (int i = 0; i < 2; ++i)
#pragma unroll
    for (int j = 0; j < 4; ++j) {
      const int col = sn * 64 + j * 16 + mr;
      const float bv = bias[col];
#pragma unroll
      for (int r = 0; r < 8; ++r)
        delta[(sm * 32 + i * 16 + r + 8 * half) * D_INNER + col] = softplus_f(acc[i][j][r] + bv);
    }
}

// ---------------- Kernel 5: selective scan (sequential over L) -------------
// Half-wave = 16 states of one channel; 2 channels/wave, 16 channels/block.
// 1-deep software pipeline + global_prefetch 16 tokens ahead (all prefetch
// targets remain inside the contiguous workspace).
__global__ __launch_bounds__(256) void k_scan(const float* __restrict__ u,
                                              const float* __restrict__ delta,
                                              const float* __restrict__ xdbl,
                                              const float* __restrict__ A_log,
                                              const float* __restrict__ Dp,
                                              float* __restrict__ y) {
  const int b      = blockIdx.x / (D_INNER / 16);
  const int dbase  = (blockIdx.x % (D_INNER / 16)) * 16;
  const int wave   = threadIdx.x >> 5;
  const int lane   = threadIdx.x & 31;
  const int n      = lane & 15;
  const int d      = dbase + wave * 2 + (lane >> 4);

  const float An = -__expf(A_log[d * D_STATE + n]);
  const float Dd = Dp[d];
  float h = 0.0f;

  const int tok0 = b * SEQ_L;
  float dt_c = delta[tok0 * D_INNER + d];
  float ut_c = u[tok0 * D_INNER + d];
  float Bv_c = xdbl[tok0 * XDBL_LD + DT_RANK + n];
  float Cv_c = xdbl[tok0 * XDBL_LD + DT_RANK + D_STATE + n];

  for (int t = 0; t < SEQ_L; ++t) {
    const int tok  = tok0 + t;
    const int tokn = tok + ((t < SEQ_L - 1) ? 1 : 0);
    // prefetch well ahead of the dependent chain (stays inside workspace)
    __builtin_prefetch((const void*)(delta + (tok + 16) * D_INNER + d), 0, 1);
    __builtin_prefetch((const void*)(u + (tok + 16) * D_INNER + d), 0, 1);
    __builtin_prefetch((const void*)(xdbl + (tok + 16) * XDBL_LD + n), 0, 1);
    // issue next-iteration loads before the exp/reduce chain
    const float dt_n = delta[tokn * D_INNER + d];
    const float ut_n = u[tokn * D_INNER + d];
    const float Bv_n = xdbl[tokn * XDBL_LD + DT_RANK + n];
    const float Cv_n = xdbl[tokn * XDBL_LD + DT_RANK + D_STATE + n];

    h = __expf(dt_c * An) * h + (dt_c * ut_c) * Bv_c;
    float p = h * Cv_c;
    p += __shfl_xor(p, 1);
    p += __shfl_xor(p, 2);
    p += __shfl_xor(p, 4);
    p += __shfl_xor(p, 8);                       // sum over the 16 states
    if (n == 0) y[tok * D_INNER + d] = p + ut_c * Dd;

    dt_c = dt_n; ut_c = ut_n; Bv_c = Bv_n; Cv_c = Cv_n;
  }
}

// ---------------- Kernel 6: y*silu(z) + LayerNorm --------------------------
__global__ __launch_bounds__(256) void k_gate_ln(const float* __restrict__ y,
                                                 const float* __restrict__ uz,
                                                 const float* __restrict__ g,
                                                 const float* __restrict__ beta,
                                                 float* __restrict__ yln) {
  const int tok = blockIdx.x;
  __shared__ float s_sum[8], s_sq[8];
  float v[3], sum = 0.0f, sq = 0.0f;
#pragma unroll
  for (int i = 0; i < 3; ++i) {
    const int d = threadIdx.x + i * 256;
    const float z   = uz[tok * N_UZ + D_INNER + d];
    const float val = y[tok * D_INNER + d] * silu_f(z);
    v[i] = val; sum += val; sq += val * val;
  }
#pragma unroll
  for (int m = 16; m >= 1; m >>= 1) { sum += __shfl_xor(sum, m); sq += __shfl_xor(sq, m); }
  const int wave = threadIdx.x >> 5, lane = threadIdx.x & 31;
  if (lane == 0) { s_sum[wave] = sum; s_sq[wave] = sq; }
  __syncthreads();
  if (wave == 0) {
    float a  = (lane < 8) ? s_sum[lane] : 0.0f;
    float b2 = (lane < 8) ? s_sq[lane]  : 0.0f;
#pragma unroll
    for (int m = 4; m >= 1; m >>= 1) { a += __shfl_xor(a, m); b2 += __shfl_xor(b2, m); }
    if (lane == 0) { s_sum[0] = a; s_sq[0] = b2; }
  }
  __syncthreads();
  const float mu  = s_sum[0] * (1.0f / D_INNER);
  const float var = s_sq[0] * (1.0f / D_INNER) - mu * mu;
  const float inv = rsqrtf(var + 1e-5f);
#pragma unroll
  for (int i = 0; i < 3; ++i) {
    const int d = threadIdx.x + i * 256;
    yln[tok * D_INNER + d] = (v[i] - mu) * inv * g[d] + beta[d];
  }
}

// ---------------- Kernel 7: out = yln @ out_proj_w + x ---------------------
__global__ __launch_bounds__(256) void k_out_proj(const float* __restrict__ yln,
                                                  const float* __restrict__ w,
                                                  const float* __restrict__ x,
                                                  float* __restrict__ out) {
  const int stripsN = D_MODEL / 64;                     // 6
  const int job = blockIdx.x * 8 + (threadIdx.x >> 5);
  const int sm = job / stripsN, sn = job % stripsN;
  v8f acc[2][4] = {};
  wmma_block<4, false>(yln + sm * 32 * D_INNER, D_INNER, w + sn * 64, D_MODEL, D_INNER, 64, acc);
  const int lane = threadIdx.x & 31;
  const int half = lane >> 4, mr = lane & 15;
#pragma unroll
  for (int i = 0; i < 2; ++i)
#pragma unroll
    for (int j = 0; j < 4; ++j) {
      const int col = sn * 64 + j * 16 + mr;
#pragma unroll
      for (int r = 0; r < 8; ++r) {
        const int row = sm * 32 + i * 16 + r + 8 * half;
        out[row * D_MODEL + col] = acc[i][j][r] + x[row * D_MODEL + col];
      }
    }
}

// ---------------------------------------------------------------------------
extern "C" void kernel_launch(void* const* d_in, const int* in_sizes, int n_in,
                              void* d_out, int out_size, void* d_ws, size_t ws_size,
                              hipStream_t stream) {
  (void)in_sizes; (void)n_in; (void)out_size; (void)ws_size;
  const float* x         = (const float*)d_in[0];
  const float* in_proj_w = (const float*)d_in[1];
  const float* conv_w    = (const float*)d_in[2];
  const float* conv_b    = (const float*)d_in[3];
  const float* x_proj_w  = (const float*)d_in[4];
  const float* dt_proj_w = (const float*)d_in[5];
  const float* dt_proj_b = (const float*)d_in[6];
  const float* A_log     = (const float*)d_in[7];
  const float* D_param   = (const float*)d_in[8];
  const float* ln_g      = (const float*)d_in[9];
  const float* ln_b      = (const float*)d_in[10];
  const float* out_proj_w= (const float*)d_in[11];
  float* out = (float*)d_out;

  // workspace layout (floats); YLN reuses U after the scan
  float* UZ   = (float*)d_ws;                       // 16384*1536
  float* U    = UZ   + (size_t)M_TOK * N_UZ;        // 16384*768
  float* XD   = U    + (size_t)M_TOK * D_INNER;     // 16384*64
  float* DT   = XD   + (size_t)M_TOK * XDBL_LD;     // 16384*768
  float* Y    = DT   + (size_t)M_TOK * D_INNER;     // 16384*768
  float* YLN  = U;

  // 1) in_proj GEMM: 512 M-strips x 24 N-strips, 8 waves/block
  k_in_proj<<<(512 * 24) / 8, 256, 0, stream>>>(x, in_proj_w, UZ);
  // 2) depthwise conv + SiLU
  k_conv<<<(M_TOK * D_INNER) / 256, 256, 0, stream>>>(UZ, conv_w, conv_b, U);
  // 3) x_proj GEMM: 512 M-strips x 1 N-strip (guarded N=56)
  k_xproj<<<512 / 8, 256, 0, stream>>>(U, x_proj_w, XD);
  // 4) dt GEMM + softplus: 512 x 12 strips
  k_dt<<<(512 * 12) / 8, 256, 0, stream>>>(XD, dt_proj_w, dt_proj_b, DT);
  // 5) selective scan: 4 batches * 48 channel-groups
  k_scan<<<BATCH * (D_INNER / 16), 256, 0, stream>>>(U, DT, XD, A_log, D_param, Y);
  // 6) gate + LayerNorm: one block per token
  k_gate_ln<<<M_TOK, 256, 0, stream>>>(Y, UZ, ln_g, ln_b, YLN);
  // 7) out_proj GEMM + residual: 512 x 6 strips
  k_out_proj<<<(512 * 6) / 8, 256, 0, stream>>>(YLN, out_proj_w, x, out);
}